// SelectiveSSM_17059610100282
// MI455X (gfx1250) — compile-verified
//
#include <hip/hip_runtime.h>
#include <math.h>

// Problem constants (from reference setup_inputs)
#define BB   4
#define LL   4096
#define DM   2048
#define SS   512
#define BL   (BB*LL)      // 16384
#define NC   64           // scan chunks per sequence
#define LC   (LL/NC)      // 64 timesteps per chunk

typedef float v2f __attribute__((ext_vector_type(2)));
typedef float v8f __attribute__((ext_vector_type(8)));

// Global-address-space pointer types: force global_load/global_store
// (LOADcnt-only path) instead of flat_* (which also ties DScnt).
typedef const float __attribute__((address_space(1)))* gcf;
typedef float       __attribute__((address_space(1)))* gf;
typedef const v2f   __attribute__((address_space(1)))* gcv2;

// ---------------------------------------------------------------------------
// Register-blocked fp32 WMMA GEMM:  C[m,n] = sum_k A[m,k] * W[n,k]
//   A : M x K row-major (lda)
//   W : N x K row-major (ldw)  (i.e. B-matrix column n = row n of W)
//   C : M x N row-major (ldc)
// Optional epilogue: C[m,n] += Dvec[n] * Xc[m,n]  (ldxc)
// Block = 128 threads = 4 waves, block tile 128(M) x 64(N), wave tile 64x32.
// K loop is software-pipelined: next k-step fragments are loaded into
// separate registers before the current 8 WMMAs issue.
// ---------------------------------------------------------------------------
__global__ __launch_bounds__(128) void gemm_wmma_f32(
    const float* __restrict__ A_, int lda,
    const float* __restrict__ W_, int ldw,
    float* __restrict__ C_, int ldc,
    int K,
    const float* __restrict__ Dvec_,
    const float* __restrict__ Xc_, int ldxc)
{
  gcf A  = (gcf)A_;
  gcf W  = (gcf)W_;
  gf  C  = (gf)C_;
  gcf Xc = (gcf)Xc_;

  const int lane  = threadIdx.x & 31;
  const int wave  = threadIdx.x >> 5;
  const int half  = lane >> 4;        // 0: lanes 0-15, 1: lanes 16-31
  const int li    = lane & 15;
  const int waveM = wave & 1;
  const int waveN = wave >> 1;
  const int m0 = blockIdx.x * 128 + waveM * 64;
  const int n0 = blockIdx.y * 64  + waveN * 32;

  v8f acc[4][2];
#pragma unroll
  for (int mt = 0; mt < 4; ++mt)
#pragma unroll
    for (int nt = 0; nt < 2; ++nt)
      acc[mt][nt] = (v8f){0.f,0.f,0.f,0.f,0.f,0.f,0.f,0.f};

  // 32-bit element offsets (max index ~33.5M floats -> fits signed i32)
  const int koff = 2 * half;
  int abase[4], bbase[2];
#pragma unroll
  for (int mt = 0; mt < 4; ++mt) abase[mt] = (m0 + mt * 16 + li) * lda + koff;
#pragma unroll
  for (int nt = 0; nt < 2; ++nt) bbase[nt] = (n0 + nt * 16 + li) * ldw + koff;

  // prologue: load k=0 fragments
  v2f a[4], b[2];
#pragma unroll
  for (int mt = 0; mt < 4; ++mt) a[mt] = *(gcv2)(A + abase[mt]);
#pragma unroll
  for (int nt = 0; nt < 2; ++nt) b[nt] = *(gcv2)(W + bbase[nt]);

#pragma unroll 2
  for (int k = 0; k < K - 4; k += 4) {
    // prefetch next k-step into distinct registers
    v2f an[4], bn[2];
#pragma unroll
    for (int mt = 0; mt < 4; ++mt) an[mt] = *(gcv2)(A + abase[mt] + k + 4);
#pragma unroll
    for (int nt = 0; nt < 2; ++nt) bn[nt] = *(gcv2)(W + bbase[nt] + k + 4);

    // compute current k-step
#pragma unroll
    for (int mt = 0; mt < 4; ++mt)
#pragma unroll
      for (int nt = 0; nt < 2; ++nt)
        acc[mt][nt] = __builtin_amdgcn_wmma_f32_16x16x4_f32(
            false, a[mt], false, b[nt], (short)0, acc[mt][nt], false, false);

#pragma unroll
    for (int mt = 0; mt < 4; ++mt) a[mt] = an[mt];
#pragma unroll
    for (int nt = 0; nt < 2; ++nt) b[nt] = bn[nt];
  }

  // epilogue k-step
#pragma unroll
  for (int mt = 0; mt < 4; ++mt)
#pragma unroll
    for (int nt = 0; nt < 2; ++nt)
      acc[mt][nt] = __builtin_amdgcn_wmma_f32_16x16x4_f32(
          false, a[mt], false, b[nt], (short)0, acc[mt][nt], false, false);

  // Store: VGPR r holds rows (r, r+8) split across half-waves, col = li
#pragma unroll
  for (int mt = 0; mt < 4; ++mt) {
#pragma unroll
    for (int nt = 0; nt < 2; ++nt) {
      const int col = n0 + nt * 16 + li;
#pragma unroll
      for (int r = 0; r < 8; ++r) {
        const int row = m0 + mt * 16 + r + 8 * half;
        float v = acc[mt][nt][r];
        if (Dvec_) v += Dvec_[col] * Xc[row * ldxc + col];
        C[row * ldc + col] = v;
      }
    }
  }
}

// ---------------------------------------------------------------------------
// Causal depthwise conv (K=4): xconv[b,l,d] = bias[d] + sum_k w[d,k]*x[b,l-3+k,d]
// Each thread produces 4 consecutive l for one d (reuses 7 loads for 4 outputs)
// ---------------------------------------------------------------------------
__global__ __launch_bounds__(256) void conv_kernel(
    const float* __restrict__ x_, const float* __restrict__ w_,
    const float* __restrict__ bias_, float* __restrict__ xconv_)
{
  gcf x = (gcf)x_;
  gcf w = (gcf)w_;
  gcf bias = (gcf)bias_;
  gf  xconv = (gf)xconv_;

  int idx = blockIdx.x * blockDim.x + threadIdx.x;   // B*(L/4)*DM threads
  int d  = idx % DM;
  int t  = idx / DM;
  int lb = t % (LL / 4);
  int b  = t / (LL / 4);
  int l0 = lb * 4;

  const int xbase = (b * LL) * DM + d;
  float xv[7];
#pragma unroll
  for (int j = 0; j < 7; ++j) {
    int l = l0 - 3 + j;
    xv[j] = (l >= 0) ? x[xbase + l * DM] : 0.f;
  }
  const float w0 = w[d * 4 + 0], w1 = w[d * 4 + 1];
  const float w2 = w[d * 4 + 2], w3 = w[d * 4 + 3];
  const float bs = bias[d];
  const int ybase = (b * LL + l0) * DM + d;
#pragma unroll
  for (int i = 0; i < 4; ++i)
    xconv[ybase + i * DM] = bs + w0 * xv[i] + w1 * xv[i + 1] + w2 * xv[i + 2] + w3 * xv[i + 3];
}

// ---------------------------------------------------------------------------
// delta column GEMV: dcol[bl] = dot(x[bl,:], Wxproj[2S,:])   (one wave per row)
// ---------------------------------------------------------------------------
__global__ __launch_bounds__(256) void dcol_kernel(
    const float* __restrict__ x_, const float* __restrict__ wrow_,
    float* __restrict__ dcol_)
{
  gcf x = (gcf)x_;
  gcf wrow = (gcf)wrow_;
  gf  dcol = (gf)dcol_;

  const int wave = threadIdx.x >> 5;
  const int lane = threadIdx.x & 31;
  const int row  = blockIdx.x * 8 + wave;
  const int xb = row * DM;
  float s = 0.f;
#pragma unroll 4
  for (int j = lane; j < DM; j += 32) s += x[xb + j] * wrow[j];
#pragma unroll
  for (int off = 16; off >= 1; off >>= 1) s += __shfl_xor(s, off, 32);
  if (lane == 0) dcol[row] = s;
}

__device__ __forceinline__ float softplus_f(float v) {
  return (v > 20.f) ? v : log1pf(expf(v));
}

// ---------------------------------------------------------------------------
// Scan phase A: per-(b,chunk) local scan -> chunk aggregate (prodA, h_end)
// thread = state index s (coalesced along s). Fuses deltaA / Bu computation.
// ---------------------------------------------------------------------------
__global__ __launch_bounds__(512) void scanA_kernel(
    const float* __restrict__ Pbc_,     // (BL,1024): [:,0:512]=Bmat
    const float* __restrict__ xstate_,  // (BL,512)
    const float* __restrict__ dcol_,    // (BL,)
    const float* __restrict__ A_log_,   // (512,)
    float* __restrict__ Aprod_, float* __restrict__ Hend_)
{
  gcf Pbc = (gcf)Pbc_;
  gcf xstate = (gcf)xstate_;
  gcf dcol = (gcf)dcol_;
  gf  Aprod = (gf)Aprod_;
  gf  Hend  = (gf)Hend_;

  const int s = threadIdx.x;
  const int c = blockIdx.x % NC;
  const int b = blockIdx.x / NC;
  const float As = -expf(A_log_[s]);
  float h = 0.f, ap = 1.f;
  const int l0 = c * LC;
  for (int i = 0; i < LC; ++i) {
    const int bl = b * LL + l0 + i;
    const float delta = softplus_f(dcol[bl]);
    const float a  = expf(As * delta);
    const float bu = delta * Pbc[bl * 1024 + s] * xstate[bl * 512 + s];
    h = a * h + bu;
    ap *= a;
  }
  const int o = (b * NC + c) * SS + s;
  Aprod[o] = ap;
  Hend[o]  = h;
}

// Scan phase B: exclusive scan over chunk aggregates -> per-chunk carry
__global__ __launch_bounds__(512) void scanB_kernel(
    const float* __restrict__ Aprod_, const float* __restrict__ Hend_,
    float* __restrict__ Hcarry_)
{
  gcf Aprod = (gcf)Aprod_;
  gcf Hend  = (gcf)Hend_;
  gf  Hcarry = (gf)Hcarry_;

  const int s = threadIdx.x;
  const int b = blockIdx.x;
  float carry = 0.f;
  for (int c = 0; c < NC; ++c) {
    const int o = (b * NC + c) * SS + s;
    Hcarry[o] = carry;
    carry = Aprod[o] * carry + Hend[o];
  }
}

// Scan phase C: replay with carry, write h_mod = Cmat*h in place over xstate
__global__ __launch_bounds__(512) void scanC_kernel(
    const float* __restrict__ Pbc_,
    float* __restrict__ xstate_,        // in: x_state, out: h_mod (aliased)
    const float* __restrict__ dcol_,
    const float* __restrict__ A_log_,
    const float* __restrict__ Hcarry_)
{
  gcf Pbc = (gcf)Pbc_;
  gf  xstate = (gf)xstate_;
  gcf dcol = (gcf)dcol_;
  gcf Hcarry = (gcf)Hcarry_;

  const int s = threadIdx.x;
  const int c = blockIdx.x % NC;
  const int b = blockIdx.x / NC;
  const float As = -expf(A_log_[s]);
  float h = Hcarry[(b * NC + c) * SS + s];
  const int l0 = c * LC;
  for (int i = 0; i < LC; ++i) {
    const int bl = b * LL + l0 + i;
    const float delta = softplus_f(dcol[bl]);
    const float a  = expf(As * delta);
    const int oS = bl * 512 + s;
    const float bu = delta * Pbc[bl * 1024 + s] * xstate[oS];
    h = a * h + bu;
    xstate[oS] = Pbc[bl * 1024 + 512 + s] * h;   // h_mod = C * h
  }
}

// ---------------------------------------------------------------------------
extern "C" void kernel_launch(void* const* d_in, const int* in_sizes, int n_in,
                              void* d_out, int out_size, void* d_ws, size_t ws_size,
                              hipStream_t stream) {
  (void)in_sizes; (void)n_in; (void)out_size; (void)ws_size;
  const float* x     = (const float*)d_in[0];  // (B,L,Dm)
  const float* Wx    = (const float*)d_in[1];  // (1025,Dm)
  const float* Win   = (const float*)d_in[2];  // (512,Dm)
  const float* Wout  = (const float*)d_in[3];  // (Dm,512)
  const float* convw = (const float*)d_in[4];  // (Dm,1,4)
  const float* convb = (const float*)d_in[5];  // (Dm,)
  const float* A_log = (const float*)d_in[6];  // (512,)
  const float* Dvec  = (const float*)d_in[7];  // (Dm,)
  float* out = (float*)d_out;

  float* ws     = (float*)d_ws;
  float* xconv  = ws;                                   // BL*DM
  float* Pbc    = xconv  + (size_t)BL * DM;             // BL*1024 (Bmat|Cmat)
  float* xstate = Pbc    + (size_t)BL * 1024;           // BL*512  (-> h_mod)
  float* dcol   = xstate + (size_t)BL * 512;            // BL
  float* Aprod  = dcol   + BL;                          // B*NC*S
  float* Hend   = Aprod  + (size_t)BB * NC * SS;
  float* Hcarry = Hend   + (size_t)BB * NC * SS;

  // 1) depthwise causal conv
  conv_kernel<<<(BB * (LL / 4) * DM) / 256, 256, 0, stream>>>(x, convw, convb, xconv);

  // 2) delta column GEMV (row 2S of Wxproj)
  dcol_kernel<<<BL / 8, 256, 0, stream>>>(x, Wx + (size_t)(2 * SS) * DM, dcol);

  // 3) ssm params GEMM: (BL x Dm) @ (Dm x 1024) -> Pbc   [B | C]
  {
    dim3 g(BL / 128, 1024 / 64);
    gemm_wmma_f32<<<g, 128, 0, stream>>>(x, DM, Wx, DM, Pbc, 1024, DM,
                                         nullptr, nullptr, 0);
  }
  // 4) x_state GEMM: (BL x Dm) @ (Dm x 512) -> xstate
  {
    dim3 g(BL / 128, SS / 64);
    gemm_wmma_f32<<<g, 128, 0, stream>>>(xconv, DM, Win, DM, xstate, SS, DM,
                                         nullptr, nullptr, 0);
  }

  // 5-7) chunked associative scan over L (fused elementwise + C-mul)
  scanA_kernel<<<BB * NC, 512, 0, stream>>>(Pbc, xstate, dcol, A_log, Aprod, Hend);
  scanB_kernel<<<BB, 512, 0, stream>>>(Aprod, Hend, Hcarry);
  scanC_kernel<<<BB * NC, 512, 0, stream>>>(Pbc, xstate, dcol, A_log, Hcarry);

  // 8) output GEMM: (BL x 512) @ (512 x Dm) + D*x_conv -> out
  {
    dim3 g(BL / 128, DM / 64);
    gemm_wmma_f32<<<g, 128, 0, stream>>>(xstate, SS, Wout, SS, out, DM, SS,
                                         Dvec, xconv, DM);
  }
}